// SoftMoEGating_30623116820827
// MI455X (gfx1250) — compile-verified
//
#include <hip/hip_runtime.h>
#include <hip/hip_bf16.h>

// SoftMoE gating for MI455X (gfx1250), fp32 end-to-end.
// Bandwidth-bound problem (~270 MB traffic -> ~12us floor @ 23.3 TB/s).
// Both GEMMs use V_WMMA_F32_16X16X4_F32 to keep fp32 precision and free VALU
// for the softmax transcendentals.

typedef float v2f __attribute__((ext_vector_type(2)));
typedef float v8f __attribute__((ext_vector_type(8)));

#define B_    8
#define S_    4096
#define D_    1024
#define KOUT  32      // NUM_EXPERTS * NUM_SLOTS
#define CHUNK 128     // tokens per workgroup in kernel 1
#define NCHUNK (S_ / CHUNK)   // 32

// Flat float offsets into d_out (reference tuple, concatenated in order)
constexpr size_t OFF_EW = 0;              // expert_weights (B,S,8)      262144
constexpr size_t OFF_EI = 262144;         // expert_indices (B,S,8)      262144
constexpr size_t OFF_PW = 524288;         // phi_weights   (B,S,8,4)    1048576
constexpr size_t OFF_SS = 1572864;        // soft_slots    (B,8,4,D)     262144
constexpr size_t OFF_XI = 1835008;        // expert_inputs (B,8,4*D)     262144
constexpr size_t OFF_PL = 2097152;        // phi_logits    (B,S,8,4)    1048576
constexpr size_t OFF_RG = 3145728;        // raw_gate_probs(B,S,8,4)    1048576

__device__ __forceinline__ v8f wmma4(v2f a, v2f b, v8f c) {
  // D(16x16 f32) = A(16x4 f32) * B(4x16 f32) + C
  return __builtin_amdgcn_wmma_f32_16x16x4_f32(
      /*neg_a=*/false, a, /*neg_b=*/false, b,
      /*c_mod=*/(short)0, c, /*reuse_a=*/false, /*reuse_b=*/false);
}

// ---------------------------------------------------------------------------
// Kernel 1: phi_logits = x @ W_phi + b  (M=tokens, N=32, K=1024)
//           + raw_gate_probs (softmax over groups of 4 columns)
//           + per-chunk online-softmax stats per (b, n) column
// grid = B * NCHUNK blocks of 256 threads (8 waves, one 16-token tile each)
// ---------------------------------------------------------------------------
__global__ void __launch_bounds__(256)
k_logits(const float* __restrict__ x, const float* __restrict__ Wp,
         const float* __restrict__ bp, float* __restrict__ out,
         float* __restrict__ wsmax, float* __restrict__ wssum) {
  const int b     = blockIdx.x >> 5;        // / NCHUNK
  const int chunk = blockIdx.x & 31;
  const int wave  = threadIdx.x >> 5;
  const int lane  = threadIdx.x & 31;
  const int half  = lane >> 4;              // 0: K{0,1} rows, 1: K{2,3} rows
  const int l16   = lane & 15;
  const int sbase = chunk * CHUNK + wave * 16;

  // A fragment source: token row (M = l16), k pair selected by half
  const float* xrow = x + ((size_t)b * S_ + sbase + l16) * D_ + half * 2;
  // B fragment source: W_phi[(k0 + 2*half) .. +1][n]
  const float* wcol = Wp + (size_t)(half * 2) * KOUT + l16;

  v8f c0 = {0.f, 0.f, 0.f, 0.f, 0.f, 0.f, 0.f, 0.f};
  v8f c1 = c0;
  for (int k0 = 0; k0 < D_; k0 += 4) {
    v2f a = *(const v2f*)(xrow + k0);                  // global_load_b64
    const float* wp0 = wcol + (size_t)k0 * KOUT;
    v2f b0 = { wp0[0],  wp0[KOUT] };                   // n tile 0..15
    v2f b1 = { wp0[16], wp0[KOUT + 16] };              // n tile 16..31
    if ((k0 & 63) == 0) __builtin_prefetch(xrow + k0 + 128, 0, 2);
    c0 = wmma4(a, b0, c0);
    c1 = wmma4(a, b1, c1);
  }

  const float bias0 = bp[l16];
  const float bias1 = bp[16 + l16];

  float l0a[8], l1a[8];
  float lm0 = -__builtin_inff(), lm1 = -__builtin_inff();
#pragma unroll
  for (int i = 0; i < 8; ++i) {
    l0a[i] = c0[i] + bias0;  lm0 = fmaxf(lm0, l0a[i]);
    l1a[i] = c1[i] + bias1;  lm1 = fmaxf(lm1, l1a[i]);
  }

  float* logits = out + OFF_PL;
  float* rgp    = out + OFF_RG;
#pragma unroll
  for (int i = 0; i < 8; ++i) {
    const int tok = sbase + i + (half << 3);           // C/D: VGPR i -> M=i / i+8
    const size_t base = ((size_t)b * S_ + tok) * KOUT;
    logits[base + l16]      = l0a[i];
    logits[base + 16 + l16] = l1a[i];
    {   // raw gate softmax over 4-lane group (same token row)
      float g = l0a[i];
      float m = fmaxf(g, __shfl_xor(g, 1)); m = fmaxf(m, __shfl_xor(m, 2));
      float e = __expf(g - m);
      float s = e + __shfl_xor(e, 1); s += __shfl_xor(s, 2);
      rgp[base + l16] = e / s;
    }
    {
      float g = l1a[i];
      float m = fmaxf(g, __shfl_xor(g, 1)); m = fmaxf(m, __shfl_xor(m, 2));
      float e = __expf(g - m);
      float s = e + __shfl_xor(e, 1); s += __shfl_xor(s, 2);
      rgp[base + 16 + l16] = e / s;
    }
  }

  // lane-local (8 rows) -> pair (16 rows) online softmax stats per column
  float s0 = 0.f, s1 = 0.f;
#pragma unroll
  for (int i = 0; i < 8; ++i) {
    s0 += __expf(l0a[i] - lm0);
    s1 += __expf(l1a[i] - lm1);
  }
  float om = __shfl_xor(lm0, 16), os = __shfl_xor(s0, 16);
  float M0 = fmaxf(lm0, om);
  float Z0 = s0 * __expf(lm0 - M0) + os * __expf(om - M0);
  om = __shfl_xor(lm1, 16); os = __shfl_xor(s1, 16);
  float M1 = fmaxf(lm1, om);
  float Z1 = s1 * __expf(lm1 - M1) + os * __expf(om - M1);

  __shared__ float smax[8][KOUT];
  __shared__ float ssm[8][KOUT];
  if (half == 0) {
    smax[wave][l16] = M0;       ssm[wave][l16] = Z0;
    smax[wave][l16 + 16] = M1;  ssm[wave][l16 + 16] = Z1;
  }
  __syncthreads();
  if (threadIdx.x < KOUT) {
    const int n = threadIdx.x;
    float M = smax[0][n], Z = ssm[0][n];
#pragma unroll
    for (int wv = 1; wv < 8; ++wv) {
      float m2 = smax[wv][n], z2 = ssm[wv][n];
      float nm = fmaxf(M, m2);
      Z = Z * __expf(M - nm) + z2 * __expf(m2 - nm);
      M = nm;
    }
    const size_t idx = ((size_t)(b * NCHUNK + chunk)) * KOUT + n;
    wsmax[idx] = M;
    wssum[idx] = Z;
  }
}

// ---------------------------------------------------------------------------
// Kernel 2: merge chunk stats -> final (max, 1/Z) per (b, n). 1 block x 256.
// ---------------------------------------------------------------------------
__global__ void __launch_bounds__(256)
k_reduce(const float* __restrict__ wsmax, const float* __restrict__ wssum,
         float* __restrict__ fmax, float* __restrict__ finvz) {
  const int t = threadIdx.x;               // 256 = B_*KOUT
  const int b = t >> 5, n = t & 31;
  float M = -__builtin_inff(), Z = 0.f;
  for (int c = 0; c < NCHUNK; ++c) {
    const size_t idx = ((size_t)(b * NCHUNK + c)) * KOUT + n;
    float m2 = wsmax[idx], z2 = wssum[idx];
    float nm = fmaxf(M, m2);
    Z = Z * __expf(M - nm) + z2 * __expf(m2 - nm);
    M = nm;
  }
  fmax[t]  = M;
  finvz[t] = 1.f / Z;
}

// ---------------------------------------------------------------------------
// Kernel 3: phi_weights = exp(l - M) / Z; constant expert_weights/indices.
// grid = B*S*KOUT/256 = 4096 blocks of 256.
// ---------------------------------------------------------------------------
__global__ void __launch_bounds__(256)
k_weights(const float* __restrict__ fmax, const float* __restrict__ finvz,
          float* __restrict__ out) {
  const size_t idx = (size_t)blockIdx.x * 256 + threadIdx.x;   // < B*S*32
  const int b = (int)(idx >> 17);          // S_*KOUT = 2^17
  const int n = (int)(idx & 31);
  const float l = out[OFF_PL + idx];
  out[OFF_PW + idx] = __expf(l - fmax[b * KOUT + n]) * finvz[b * KOUT + n];
  if (n < 8) {                             // (B,S,E) planes
    const size_t bs = idx >> 5;
    out[OFF_EW + bs * 8 + n] = 0.125f;
    out[OFF_EI + bs * 8 + n] = (float)n;
  }
}

// ---------------------------------------------------------------------------
// Kernel 4: soft_slots[b, n, d] = sum_s phi_weights[b,s,n] * x[b,s,d]
//           (M=32 slots, N=d, K=s), also duplicated into expert_inputs.
// grid = B * (D_/128) = 64 blocks of 256 (8 waves; one 16-wide d tile each).
// ---------------------------------------------------------------------------
__global__ void __launch_bounds__(256)
k_slots(const float* __restrict__ x, float* __restrict__ out) {
  const float* pw = out + OFF_PW;
  float* ss = out + OFF_SS;
  float* xi = out + OFF_XI;
  const int b      = blockIdx.x >> 3;
  const int dchunk = blockIdx.x & 7;
  const int wave = threadIdx.x >> 5;
  const int lane = threadIdx.x & 31;
  const int half = lane >> 4;
  const int l16  = lane & 15;
  const int d0   = dchunk * 128 + wave * 16;

  // A: phi_weights^T -> A[m=slot, k=s]; half selects K rows {0,1} vs {2,3}
  const float* wrow = pw + ((size_t)b * S_ + half * 2) * KOUT;
  // B: x rows -> B[k=s, n=d]
  const float* xcol = x + ((size_t)b * S_ + half * 2) * D_ + d0 + l16;

  v8f cA = {0.f, 0.f, 0.f, 0.f, 0.f, 0.f, 0.f, 0.f};
  v8f cB = cA;
  for (int s0 = 0; s0 < S_; s0 += 4) {
    const float* wp = wrow + (size_t)s0 * KOUT;
    v2f a0 = { wp[l16],      wp[KOUT + l16] };        // slots 0..15
    v2f a1 = { wp[16 + l16], wp[KOUT + 16 + l16] };   // slots 16..31
    const float* xp = xcol + (size_t)s0 * D_;
    v2f bb = { xp[0], xp[D_] };
    if ((s0 & 63) == 0) __builtin_prefetch(xp + 4 * D_, 0, 2);
    cA = wmma4(a0, bb, cA);
    cB = wmma4(a1, bb, cB);
  }

#pragma unroll
  for (int i = 0; i < 8; ++i) {
    const int slot = i + (half << 3);
    const size_t fl  = ((size_t)b * KOUT + slot) * D_ + d0 + l16;
    ss[fl] = cA[i];  xi[fl] = cA[i];
    const size_t fl2 = ((size_t)b * KOUT + slot + 16) * D_ + d0 + l16;
    ss[fl2] = cB[i]; xi[fl2] = cB[i];
  }
}

// ---------------------------------------------------------------------------
extern "C" void kernel_launch(void* const* d_in, const int* in_sizes, int n_in,
                              void* d_out, int out_size, void* d_ws, size_t ws_size,
                              hipStream_t stream) {
  const float* x  = (const float*)d_in[0];   // (B,S,D)
  const float* Wp = (const float*)d_in[1];   // (D,32)
  const float* bp = (const float*)d_in[2];   // (32,)
  float* out = (float*)d_out;
  float* ws  = (float*)d_ws;

  float* wsmax = ws;            // B*NCHUNK*KOUT = 8192 floats
  float* wssum = ws + 8192;     // 8192 floats
  float* fmax  = ws + 16384;    // 256 floats
  float* finvz = ws + 16640;    // 256 floats

  k_logits <<<B_ * NCHUNK, 256, 0, stream>>>(x, Wp, bp, out, wsmax, wssum);
  k_reduce <<<1, 256, 0, stream>>>(wsmax, wssum, fmax, finvz);
  k_weights<<<(B_ * S_ * KOUT) / 256, 256, 0, stream>>>(fmax, finvz, out);
  k_slots  <<<B_ * (D_ / 128), 256, 0, stream>>>(x, out);
}